// GCN_17377437679657
// MI455X (gfx1250) — compile-verified
//
#include <hip/hip_runtime.h>
#include <math.h>

#define N_NODES 100000
#define IN_CH   128
#define HID_CH  64
#define OUT_CH  8
#define N_EDGES 3200000

typedef __attribute__((ext_vector_type(2))) float v2f;
typedef __attribute__((ext_vector_type(8))) float v8f;

// ---------------- init / degree kernels ----------------

__global__ void zero_f32(float* __restrict__ p, int n) {
  int i = blockIdx.x * blockDim.x + threadIdx.x;
  if (i < n) p[i] = 0.0f;
}

__global__ void init_deg(float* __restrict__ deg) {
  int i = blockIdx.x * blockDim.x + threadIdx.x;
  if (i < N_NODES) deg[i] = 1.0f;  // self-loop contribution
}

__global__ void accum_deg(const long long* __restrict__ dst, float* __restrict__ deg) {
  int e = blockIdx.x * blockDim.x + threadIdx.x;
  if (e < N_EDGES) atomicAdd(&deg[(int)dst[e]], 1.0f);
}

__global__ void deg_to_dinv(float* __restrict__ deg) {
  int i = blockIdx.x * blockDim.x + threadIdx.x;
  if (i < N_NODES) deg[i] = rsqrtf(deg[i]);  // deg >= 1 always
}

__global__ void pad_w2(const float* __restrict__ W2, float* __restrict__ W2p) {
  int i = blockIdx.x * blockDim.x + threadIdx.x;  // 64*16 entries
  if (i < HID_CH * 16) {
    int k = i >> 4, n = i & 15;
    W2p[i] = (n < OUT_CH) ? W2[k * OUT_CH + n] : 0.0f;
  }
}

// ---------------- WMMA GEMMs (fp32, 16x16x4) ----------------
// Layout per CDNA5 ISA 7.12.2:
//  A 16x4: lanes 0-15 -> K=0(V0),1(V1); lanes 16-31 -> K=2(V0),3(V1); M = lane&15
//  B 4x16: lanes 0-15 -> K=0(V0),1(V1); lanes 16-31 -> K=2(V0),3(V1); N = lane&15
//  C/D 16x16: VGPR v, M = v + 8*(lane>=16), N = lane&15

// h1 = X[100000,128] @ W1[128,64]; 4 waves/block, each wave owns one 16-col tile.
__global__ __launch_bounds__(128) void gemm1_wmma(const float* __restrict__ X,
                                                  const float* __restrict__ W,
                                                  float* __restrict__ H) {
  const int lane = threadIdx.x & 31;
  const int wave = threadIdx.x >> 5;         // 0..3 -> N tile
  const int rowBase = blockIdx.x * 16;       // 6250 tiles exactly
  const int l15  = lane & 15;
  const int kOff = (lane >> 4) * 2;          // 0 or 2
  const int nCol = wave * 16 + l15;
  const float* xr = X + (size_t)(rowBase + l15) * IN_CH;
  v8f acc = {};
#pragma unroll 4
  for (int k = 0; k < IN_CH; k += 4) {
    v2f a, b;
    a.x = xr[k + kOff];
    a.y = xr[k + kOff + 1];
    b.x = W[(k + kOff) * HID_CH + nCol];
    b.y = W[(k + kOff + 1) * HID_CH + nCol];
    acc = __builtin_amdgcn_wmma_f32_16x16x4_f32(false, a, false, b,
                                                (short)0, acc, false, false);
  }
  const int rOut = rowBase + (lane >> 4) * 8;
#pragma unroll
  for (int v = 0; v < 8; ++v)
    H[(size_t)(rOut + v) * HID_CH + nCol] = acc[v];
}

// h2 = Hin[100000,64] @ W2p[64,16] (cols 8..15 are zero padding);
// 2 waves/block, each wave owns one 16-row tile; store only cols < 8.
__global__ __launch_bounds__(64) void gemm2_wmma(const float* __restrict__ Hin,
                                                 const float* __restrict__ W2p,
                                                 float* __restrict__ H2) {
  const int lane = threadIdx.x & 31;
  const int wave = threadIdx.x >> 5;
  const int rowBase = (blockIdx.x * 2 + wave) * 16;  // 6250 tiles exactly
  const int l15  = lane & 15;
  const int kOff = (lane >> 4) * 2;
  const float* hr = Hin + (size_t)(rowBase + l15) * HID_CH;
  v8f acc = {};
#pragma unroll 4
  for (int k = 0; k < HID_CH; k += 4) {
    v2f a, b;
    a.x = hr[k + kOff];
    a.y = hr[k + kOff + 1];
    b.x = W2p[(k + kOff) * 16 + l15];
    b.y = W2p[(k + kOff + 1) * 16 + l15];
    acc = __builtin_amdgcn_wmma_f32_16x16x4_f32(false, a, false, b,
                                                (short)0, acc, false, false);
  }
  if (l15 < OUT_CH) {
    const int rOut = rowBase + (lane >> 4) * 8;
#pragma unroll
    for (int v = 0; v < 8; ++v)
      H2[(size_t)(rOut + v) * OUT_CH + l15] = acc[v];
  }
}

// ---------------- edge aggregation (bandwidth/atomic bound) ----------------

// 16 threads per edge, 4 channels each (float4 gather, 4 f32 atomics)
__global__ void scatter1(const long long* __restrict__ src,
                         const long long* __restrict__ dst,
                         const float* __restrict__ dinv,
                         const float* __restrict__ H,
                         float* __restrict__ agg) {
  int idx = blockIdx.x * blockDim.x + threadIdx.x;   // E*16 = 51.2M threads
  int e  = idx >> 4;
  int cq = (idx & 15) * 4;
  if (e < N_EDGES) {
    int s = (int)src[e], d = (int)dst[e];
    float nrm = dinv[s] * dinv[d];
    const float4 hv = *(const float4*)(H + (size_t)s * HID_CH + cq);
    float* ap = agg + (size_t)d * HID_CH + cq;
    atomicAdd(ap + 0, nrm * hv.x);
    atomicAdd(ap + 1, nrm * hv.y);
    atomicAdd(ap + 2, nrm * hv.z);
    atomicAdd(ap + 3, nrm * hv.w);
  }
}

// 2 threads per edge, 4 channels each
__global__ void scatter2(const long long* __restrict__ src,
                         const long long* __restrict__ dst,
                         const float* __restrict__ dinv,
                         const float* __restrict__ H2,
                         float* __restrict__ agg2) {
  int idx = blockIdx.x * blockDim.x + threadIdx.x;   // E*2 threads
  int e  = idx >> 1;
  int cq = (idx & 1) * 4;
  if (e < N_EDGES) {
    int s = (int)src[e], d = (int)dst[e];
    float nrm = dinv[s] * dinv[d];
    const float4 hv = *(const float4*)(H2 + (size_t)s * OUT_CH + cq);
    float* ap = agg2 + (size_t)d * OUT_CH + cq;
    atomicAdd(ap + 0, nrm * hv.x);
    atomicAdd(ap + 1, nrm * hv.y);
    atomicAdd(ap + 2, nrm * hv.z);
    atomicAdd(ap + 3, nrm * hv.w);
  }
}

// agg = relu(agg + dinv^2 * H + b1)   (self-loop term + bias + relu, in place)
__global__ void finalize1(const float* __restrict__ H, const float* __restrict__ dinv,
                          const float* __restrict__ b1, float* __restrict__ agg) {
  int idx = blockIdx.x * blockDim.x + threadIdx.x;   // N*64
  if (idx < N_NODES * HID_CH) {
    int i = idx >> 6, c = idx & 63;
    float di = dinv[i];
    float v = agg[idx] + di * di * H[idx] + b1[c];
    agg[idx] = fmaxf(v, 0.0f);
  }
}

// out = log_softmax(agg2 + dinv^2 * H2 + b2) over 8 channels
__global__ void finalize2_lsm(const float* __restrict__ H2, const float* __restrict__ dinv,
                              const float* __restrict__ b2, const float* __restrict__ agg2,
                              float* __restrict__ out) {
  int i = blockIdx.x * blockDim.x + threadIdx.x;
  if (i < N_NODES) {
    float di = dinv[i];
    float di2 = di * di;
    float4 a0 = *(const float4*)(agg2 + (size_t)i * 8);
    float4 a1 = *(const float4*)(agg2 + (size_t)i * 8 + 4);
    float4 h0 = *(const float4*)(H2 + (size_t)i * 8);
    float4 h1 = *(const float4*)(H2 + (size_t)i * 8 + 4);
    float v[8];
    v[0] = a0.x + di2 * h0.x + b2[0];
    v[1] = a0.y + di2 * h0.y + b2[1];
    v[2] = a0.z + di2 * h0.z + b2[2];
    v[3] = a0.w + di2 * h0.w + b2[3];
    v[4] = a1.x + di2 * h1.x + b2[4];
    v[5] = a1.y + di2 * h1.y + b2[5];
    v[6] = a1.z + di2 * h1.z + b2[6];
    v[7] = a1.w + di2 * h1.w + b2[7];
    float m = v[0];
#pragma unroll
    for (int c = 1; c < 8; ++c) m = fmaxf(m, v[c]);
    float s = 0.0f;
#pragma unroll
    for (int c = 0; c < 8; ++c) s += __expf(v[c] - m);
    float lse = m + __logf(s);
    float4 o0 = make_float4(v[0] - lse, v[1] - lse, v[2] - lse, v[3] - lse);
    float4 o1 = make_float4(v[4] - lse, v[5] - lse, v[6] - lse, v[7] - lse);
    *(float4*)(out + (size_t)i * 8)     = o0;
    *(float4*)(out + (size_t)i * 8 + 4) = o1;
  }
}

// ---------------- driver ----------------

extern "C" void kernel_launch(void* const* d_in, const int* in_sizes, int n_in,
                              void* d_out, int out_size, void* d_ws, size_t ws_size,
                              hipStream_t stream) {
  const float*      x  = (const float*)d_in[0];
  const long long*  ei = (const long long*)d_in[1];   // int64 edge_index [2, E]
  const float*      W1 = (const float*)d_in[2];
  const float*      b1 = (const float*)d_in[3];
  const float*      W2 = (const float*)d_in[4];
  const float*      b2 = (const float*)d_in[5];
  const long long* src = ei;
  const long long* dst = ei + N_EDGES;
  float* out = (float*)d_out;

  // workspace layout (~58 MB total)
  float* ws   = (float*)d_ws;
  float* dinv = ws;                                  // N        (deg then rsqrt, in place)
  float* h1   = dinv + N_NODES;                      // N*64
  float* agg1 = h1 + (size_t)N_NODES * HID_CH;       // N*64
  float* h2   = agg1 + (size_t)N_NODES * HID_CH;     // N*8
  float* agg2 = h2 + (size_t)N_NODES * OUT_CH;       // N*8
  float* w2p  = agg2 + (size_t)N_NODES * OUT_CH;     // 64*16

  init_deg<<<(N_NODES + 255) / 256, 256, 0, stream>>>(dinv);
  zero_f32<<<(N_NODES * HID_CH + 255) / 256, 256, 0, stream>>>(agg1, N_NODES * HID_CH);
  zero_f32<<<(N_NODES * OUT_CH + 255) / 256, 256, 0, stream>>>(agg2, N_NODES * OUT_CH);
  pad_w2<<<4, 256, 0, stream>>>(W2, w2p);

  accum_deg<<<(N_EDGES + 255) / 256, 256, 0, stream>>>(dst, dinv);
  deg_to_dinv<<<(N_NODES + 255) / 256, 256, 0, stream>>>(dinv);

  gemm1_wmma<<<N_NODES / 16, 128, 0, stream>>>(x, W1, h1);                 // 6250 blocks
  scatter1<<<(N_EDGES * 16) / 256, 256, 0, stream>>>(src, dst, dinv, h1, agg1); // 200000
  finalize1<<<(N_NODES * HID_CH + 255) / 256, 256, 0, stream>>>(h1, dinv, b1, agg1);

  gemm2_wmma<<<N_NODES / 32, 64, 0, stream>>>(agg1, w2p, h2);              // 3125 blocks
  scatter2<<<(N_EDGES * 2) / 256, 256, 0, stream>>>(src, dst, dinv, h2, agg2);  // 25000
  finalize2_lsm<<<(N_NODES + 255) / 256, 256, 0, stream>>>(h2, dinv, b2, agg2, out);
}